// DifferNet_FIT_51427938402584
// MI455X (gfx1250) — compile-verified
//
#include <hip/hip_runtime.h>
#include <hip/hip_bf16.h>

// ---------------------------------------------------------------------------
// Strain pipeline for MI455X (gfx1250), wave32.
// Memory-bound (~28 GFLOP vs ~0.7 GB @ 23.3 TB/s) -> fp32 end-to-end.
// Convs = implicit GEMM on v_wmma_f32_16x16x4_f32 with LDS-staged input tiles
// (async global->LDS via global_load_async_to_lds_b128) and LDS-resident
// packed weights. A and B operands each come from one ds_load_b64 per WMMA.
// ---------------------------------------------------------------------------

#define HH 1536
#define WW 1536
#define HWF (HH * WW)           // 2359296
#define H2 768
#define W2 768
#define HW2F (H2 * W2)          // 589824

typedef float v2f __attribute__((ext_vector_type(2)));
typedef float v8f __attribute__((ext_vector_type(8)));
typedef int   v4i_b128 __attribute__((vector_size(16)));   // matches builtin param

#if __has_builtin(__builtin_amdgcn_global_load_async_to_lds_b128)
#define HAVE_ASYNC_LDS 1
#else
#define HAVE_ASYNC_LDS 0
#endif

__device__ __forceinline__ void wait_async_then_barrier() {
#if HAVE_ASYNC_LDS
#if __has_builtin(__builtin_amdgcn_s_wait_asynccnt)
    __builtin_amdgcn_s_wait_asynccnt(0);
#else
    asm volatile("s_wait_asynccnt 0x0" ::: "memory");
#endif
#endif
    __syncthreads();
}

// ---------------------------------------------------------------------------
// Pack OIHW weights (Cout, Cin, Kh, Kw) into the WMMA B-operand lane layout:
// wp[((c*32 + lane)*2) + j] = W[k][n],  k = c*4 + (lane>>4)*2 + j, n = lane&15
// K ordering: k = (kh*Kw + kw)*Cin + cin  (cin fastest -> contiguous in NHWC)
// ---------------------------------------------------------------------------
__global__ void pack_weights(const float* __restrict__ w, float* __restrict__ wp,
                             int Cout, int Cin, int Kh, int Kw) {
    int K  = Cin * Kh * Kw;
    int K4 = (K + 3) >> 2;
    int idx = blockIdx.x * blockDim.x + threadIdx.x;
    int total = K4 * 64;
    if (idx >= total) return;
    int j    = idx & 1;
    int lane = (idx >> 1) & 31;
    int c    = idx >> 6;
    int n    = lane & 15;
    int k    = c * 4 + ((lane >> 4) << 1) + j;
    float v = 0.0f;
    if (k < K && n < Cout) {
        int tap = k / Cin, cin = k - tap * Cin;
        int kh = tap / Kw, kw = tap - kh * Kw;
        v = w[((n * Cin + cin) * Kh + kh) * Kw + kw];
    }
    wp[idx] = v;
}

// NCHW (1,2,H,W) -> NHWC2
__global__ void to_nhwc2(const float* __restrict__ in, float* __restrict__ out) {
    int p = blockIdx.x * blockDim.x + threadIdx.x;
    if (p < HWF) {
        out[p * 2 + 0] = in[p];
        out[p * 2 + 1] = in[HWF + p];
    }
}

// ---------------------------------------------------------------------------
// Implicit-GEMM conv via V_WMMA_F32_16X16X4_F32.
// Block = NW waves, covers NW*16 consecutive output pixels of ONE output row
// (Wout % (NW*16) == 0 for all layers). Input halo tile staged into LDS
// (async global->LDS on interior tiles), weights staged into LDS.
// pad = Kh/2, Kw/2 (matches all layers in this net).
// ---------------------------------------------------------------------------
template <int Kh, int Kw, int Cin, int STRIDE, int NW, bool RELU, bool OUT_NCHW>
__global__ __launch_bounds__(32 * NW)
void conv_wmma(const float* __restrict__ gin,     // NHWC, Cin channels
               const float* __restrict__ wp,      // packed B layout [K4][32][2]
               const float* __restrict__ bias,
               float* __restrict__ out,
               int Hin, int Win, int Hout, int Wout,
               int cout_stride, int ch_off, int cout_write) {
    constexpr int K    = Cin * Kh * Kw;
    constexpr int K4   = (K + 3) / 4;
    constexpr int PADH = Kh / 2;
    constexpr int PADW = Kw / 2;
    constexpr int TLW  = (NW * 16 - 1) * STRIDE + Kw;  // staged input width
    constexpr int TLWC = TLW * Cin;                    // floats per staged row
    constexpr int NT   = 32 * NW;
    static_assert(TLWC % 4 == 0, "b128 staging requires TLWC % 4 == 0");

    __shared__ __align__(16) float sIn[Kh * TLWC];
    __shared__ __align__(16) float sW[K4 * 64];

    const int tid = threadIdx.y * 32 + threadIdx.x;

    // ---- weights -> LDS ----
    for (int i = tid; i < K4 * 64; i += NT) sW[i] = wp[i];

    // ---- input halo tile -> LDS (async on interior tiles) ----
    const int pb  = blockIdx.x * (NW * 16);   // first output pixel of block
    const int y0  = pb / Wout;
    const int x0  = pb - y0 * Wout;
    const int ix0 = x0 * STRIDE - PADW;
    const int iy0 = y0 * STRIDE - PADH;

#pragma unroll
    for (int r = 0; r < Kh; ++r) {
        const int  iy   = iy0 + r;
        const bool iyOK = (iy >= 0) && (iy < Hin);
        float* lrow = &sIn[r * TLWC];
        if (iyOK && ix0 >= 0 && (ix0 + TLW) <= Win) {      // block-uniform
            const float* g = gin + ((size_t)iy * Win + ix0) * Cin;
#if HAVE_ASYNC_LDS
            for (int t4 = tid; t4 < TLWC / 4; t4 += NT)
                __builtin_amdgcn_global_load_async_to_lds_b128(
                    (__attribute__((address_space(1))) v4i_b128*)(g + t4 * 4),
                    (__attribute__((address_space(3))) v4i_b128*)(lrow + t4 * 4),
                    0, 0);
#else
            for (int t4 = tid; t4 < TLWC / 4; t4 += NT)
                *(float4*)(lrow + t4 * 4) = *(const float4*)(g + t4 * 4);
#endif
        } else {                                            // edge: zero-fill
            for (int t = tid; t < TLWC; t += NT) {
                int xx = t / Cin, ci = t - xx * Cin;
                int ix = ix0 + xx;
                float v = 0.0f;
                if (iyOK && ix >= 0 && ix < Win)
                    v = gin[((size_t)iy * Win + ix) * Cin + ci];
                lrow[t] = v;
            }
        }
    }
    wait_async_then_barrier();

    // ---- GEMM: one wave = 16 pixels x 16 channels ----
    const int lane  = threadIdx.x;
    const int wv    = threadIdx.y;
    const int m     = lane & 15;
    const int khalf = lane >> 4;          // lanes 16-31 carry K+2,K+3
    const int lp    = wv * 16 + m;        // local pixel (A-matrix row)

    v8f acc = {};
    for (int c = 0; c < K4; ++c) {
        const int k0 = c * 4 + khalf * 2;
        v2f a = {0.0f, 0.0f};
        if (k0 < K) {
            int tap = k0 / Cin, cin = k0 - tap * Cin;
            int kh = tap / Kw, kw = tap - kh * Kw;
            a = *(const v2f*)&sIn[(kh * TLW + lp * STRIDE + kw) * Cin + cin];
        }
        v2f b = *(const v2f*)&sW[(c * 32 + lane) * 2];
        acc = __builtin_amdgcn_wmma_f32_16x16x4_f32(
            false, a, false, b, (short)0, acc, false, false);
    }

    // C/D layout: lane l, vgpr r -> M = r + 8*(l>>4), N = l&15
    const int n = lane & 15;
    if (n < cout_write) {
        float bn = bias[n];
#pragma unroll
        for (int r = 0; r < 8; ++r) {
            int pp = pb + wv * 16 + r + khalf * 8;
            float v = acc[r] + bn;
            if (RELU) v = fmaxf(v, 0.0f);
            if (OUT_NCHW)
                out[n * (Hout * Wout) + pp] = v;
            else
                out[(size_t)pp * cout_stride + ch_off + n] = v;
        }
    }
}

// ---------------------------------------------------------------------------
// ConvTranspose2d 16->8, k=3, stride=2, pad=1, out_pad=1 (+ReLU).
// Output parity selects <=4 taps. Input NHWC16 @768^2, output NHWC8 @1536^2.
// ---------------------------------------------------------------------------
__global__ void convT_kernel(const float* __restrict__ in,
                             const float* __restrict__ wt,   // (16,8,3,3)
                             const float* __restrict__ bt,
                             float* __restrict__ out) {
    int p = blockIdx.x * blockDim.x + threadIdx.x;
    if (p >= HWF) return;
    int y = p / WW, x = p - y * WW;
    float acc[8];
#pragma unroll
    for (int o = 0; o < 8; ++o) acc[o] = bt[o];

#pragma unroll
    for (int kh = 0; kh < 3; ++kh) {
        int ty = y + 1 - kh;
        if (ty & 1) continue;
        int iy = ty >> 1;
        if (iy < 0 || iy >= H2) continue;
#pragma unroll
        for (int kw = 0; kw < 3; ++kw) {
            int tx = x + 1 - kw;
            if (tx & 1) continue;
            int ix = tx >> 1;
            if (ix < 0 || ix >= W2) continue;
            const float* ip = in + ((size_t)iy * W2 + ix) * 16;
            for (int cin = 0; cin < 16; ++cin) {
                float v = ip[cin];
                const float* wrow = wt + cin * 72 + kh * 3 + kw;
#pragma unroll
                for (int o = 0; o < 8; ++o)
                    acc[o] = fmaf(v, wrow[o * 9], acc[o]);
            }
        }
    }
    float* op = out + (size_t)p * 8;
#pragma unroll
    for (int o = 0; o < 8; ++o) op[o] = fmaxf(acc[o], 0.0f);
}

// ---------------------------------------------------------------------------
// Separable raw strain. kx = c_j/6864 (rank-1), ky = c_i/6864 (rank-1), so the
// 12x12 dilated correlation factors into an x-pass + y-pass (~6x fewer loads).
// (*100 then /100 in the reference cancels.)
// ---------------------------------------------------------------------------
__global__ void strain_pass1(const float* __restrict__ disp,
                             float* __restrict__ tbuf) {
    int p = blockIdx.x * blockDim.x + threadIdx.x;
    if (p >= HWF) return;
    int y = p / WW, x = p - y * WW;
    const float* rx = disp + (size_t)y * WW;
    const float* ry = rx + HWF;
    float t0 = 0.f, t1 = 0.f, t2 = 0.f, t3 = 0.f;
#pragma unroll
    for (int j = 0; j < 12; ++j) {
        int cj = 2 * j - 11;
        int xx = x + cj;
        if (xx < 0 || xx >= WW) continue;
        float fcj = (float)cj;
        float ux = rx[xx], uy = ry[xx];
        t0 = fmaf(fcj, ux, t0);  t1 += ux;
        t2 = fmaf(fcj, uy, t2);  t3 += uy;
    }
    float4* o = (float4*)(tbuf + (size_t)p * 4);
    *o = make_float4(t0, t1, t2, t3);
}

__global__ void strain_pass2(const float* __restrict__ tbuf,
                             float* __restrict__ concat8) {
    int p = blockIdx.x * blockDim.x + threadIdx.x;
    if (p >= HWF) return;
    int y = p / WW, x = p - y * WW;
    const float inv = 1.0f / 6864.0f;
    float exx = 0.f, exy = 0.f, eyx = 0.f, eyy = 0.f;
#pragma unroll
    for (int i = 0; i < 12; ++i) {
        int ci = 2 * i - 11;
        int yy = y + ci;
        if (yy < 0 || yy >= HH) continue;
        const float* tp = tbuf + ((size_t)yy * WW + x) * 4;
        __builtin_prefetch(tp, 0, 0);            // global_prefetch_b8
        float4 t = *(const float4*)tp;
        float fci = (float)ci;
        exx += t.x;                // sum_i [sum_j c_j ux]
        exy = fmaf(fci, t.y, exy); // sum_i c_i [sum_j ux]
        eyx += t.z;
        eyy = fmaf(fci, t.w, eyy);
    }
    float* o = concat8 + (size_t)p * 8 + 4;
    o[0] = exx * inv; o[1] = exy * inv; o[2] = eyx * inv; o[3] = eyy * inv;
}

// ---------------------------------------------------------------------------
// Host-side orchestration (graph-capture safe: only kernel launches on stream)
// ---------------------------------------------------------------------------
extern "C" void kernel_launch(void* const* d_in, const int* in_sizes, int n_in,
                              void* d_out, int out_size, void* d_ws, size_t ws_size,
                              hipStream_t stream) {
    (void)in_sizes; (void)n_in; (void)out_size; (void)ws_size;

    const float* disp = (const float*)d_in[0];
    const float* w1 = (const float*)d_in[1];  const float* b1 = (const float*)d_in[2];
    const float* w2 = (const float*)d_in[3];  const float* b2 = (const float*)d_in[4];
    const float* w3 = (const float*)d_in[5];  const float* b3 = (const float*)d_in[6];
    const float* w4 = (const float*)d_in[7];  const float* b4 = (const float*)d_in[8];
    const float* wt = (const float*)d_in[9];  const float* bt = (const float*)d_in[10];
    const float* w5 = (const float*)d_in[11]; const float* b5 = (const float*)d_in[12];
    const float* wc = (const float*)d_in[13]; const float* bc = (const float*)d_in[14];
    float* out = (float*)d_out;
    float* ws  = (float*)d_ws;

    // ---- workspace carve-up (floats) ----
    size_t o = 0;
    float* nhwc2 = ws + o; o += (size_t)2  * HWF;   // conv1 input
    float* bufA  = ws + o; o += (size_t)16 * HWF;   // act1; later act4 + actT
    float* bufB  = ws + o; o += (size_t)16 * HW2F;  // act2; later concat head
    float* bufC  = ws + o; o += (size_t)16 * HW2F;  // act3; concat tail
    float* tbuf  = ws + o; o += (size_t)4  * HWF;   // strain separable interm.
    float* wp1 = ws + o; o += 13  * 64;
    float* wp2 = ws + o; o += 100 * 64;
    float* wp3 = ws + o; o += 100 * 64;
    float* wp4 = ws + o; o += 36  * 64;
    float* wp5 = ws + o; o += 18  * 64;
    float* wpc = ws + o; o += 18  * 64;

    float* act1    = bufA;
    float* act2    = bufB;
    float* act3    = bufC;
    float* act4    = bufA;                       // act1 dead after conv2
    float* actT    = bufA + (size_t)8 * HWF;     // disjoint from act4 (4*HWF)
    float* concat8 = bufB;                       // bufB+bufC = exactly 8*HWF

    // ---- pack weights into WMMA B lane layout ----
    auto pk = [&](const float* w, float* wp, int Cout, int Cin, int Kh, int Kw) {
        int total = (((Cin * Kh * Kw) + 3) >> 2) * 64;
        pack_weights<<<(total + 255) / 256, 256, 0, stream>>>(w, wp, Cout, Cin, Kh, Kw);
    };
    pk(w1, wp1, 16, 2, 5, 5);
    pk(w2, wp2, 16, 16, 5, 5);
    pk(w3, wp3, 16, 16, 5, 5);
    pk(w4, wp4, 16, 16, 3, 3);
    pk(w5, wp5, 4, 8, 3, 3);
    pk(wc, wpc, 4, 8, 3, 3);

    // ---- input layout ----
    to_nhwc2<<<(HWF + 255) / 256, 256, 0, stream>>>(disp, nhwc2);

    // ---- CNN trunk (implicit GEMM via WMMA; LDS-staged tiles) ----
    conv_wmma<5, 5, 2, 1, 8, true, false><<<HWF / 128, dim3(32, 8), 0, stream>>>(
        nhwc2, wp1, b1, act1, HH, WW, HH, WW, 16, 0, 16);
    conv_wmma<5, 5, 16, 2, 2, true, false><<<HW2F / 32, dim3(32, 2), 0, stream>>>(
        act1, wp2, b2, act2, HH, WW, H2, W2, 16, 0, 16);
    conv_wmma<5, 5, 16, 1, 2, true, false><<<HW2F / 32, dim3(32, 2), 0, stream>>>(
        act2, wp3, b3, act3, H2, W2, H2, W2, 16, 0, 16);
    conv_wmma<3, 3, 16, 1, 8, true, false><<<HW2F / 128, dim3(32, 8), 0, stream>>>(
        act3, wp4, b4, act4, H2, W2, H2, W2, 16, 0, 16);

    // ---- transposed conv (parity-decomposed) ----
    convT_kernel<<<(HWF + 255) / 256, 256, 0, stream>>>(act4, wt, bt, actT);

    // ---- residue conv (8->4, no relu) into concat channels 0..3 ----
    conv_wmma<3, 3, 8, 1, 8, false, false><<<HWF / 128, dim3(32, 8), 0, stream>>>(
        actT, wp5, b5, concat8, HH, WW, HH, WW, 8, 0, 4);

    // ---- separable raw strain into concat channels 4..7 ----
    strain_pass1<<<(HWF + 255) / 256, 256, 0, stream>>>(disp, tbuf);
    strain_pass2<<<(HWF + 255) / 256, 256, 0, stream>>>(tbuf, concat8);

    // ---- final fuse conv (8->4) -> NCHW output ----
    conv_wmma<3, 3, 8, 1, 8, false, true><<<HWF / 128, dim3(32, 8), 0, stream>>>(
        concat8, wpc, bc, out, HH, WW, HH, WW, 8, 0, 4);
}